// HDLUT_87454124081250
// MI455X (gfx1250) — compile-verified
//
#include <hip/hip_runtime.h>
#include <stdint.h>

#define NN 4096
#define ON 8192

typedef __attribute__((ext_vector_type(4))) unsigned int v4u;
typedef __attribute__((ext_vector_type(8))) int         v8i;
typedef __attribute__((ext_vector_type(4))) int         v4i;
typedef __attribute__((ext_vector_type(4))) float       v4f;

// 8 LDS float4 gathers + compile-time slot permutations (rotation inverses).
__device__ __forceinline__ v4f gather8(const v4f* lut, int c,
                                       int nE, int nS, int nW, int nN,
                                       int nSE, int nSW, int nNW, int nNE) {
  const v4f* L0 = lut;        // kt = 0 (axial)
  const v4f* L1 = lut + 256;  // kt = 1 (diagonal)
  const int b = c << 4;       // c * 16
  v4f a = {0.0f, 0.0f, 0.0f, 0.0f};
  v4f g;
  g = L0[b + nE];  a.x += g.x; a.y += g.y; a.z += g.z; a.w += g.w; // r=0 id
  g = L0[b + nS];  a.x += g.z; a.y += g.x; a.z += g.w; a.w += g.y; // r=1
  g = L0[b + nW];  a.x += g.w; a.y += g.z; a.z += g.y; a.w += g.x; // r=2 rev
  g = L0[b + nN];  a.x += g.y; a.y += g.w; a.z += g.x; a.w += g.z; // r=3
  g = L1[b + nSE]; a.x += g.x; a.y += g.y; a.z += g.z; a.w += g.w;
  g = L1[b + nSW]; a.x += g.z; a.y += g.x; a.z += g.w; a.w += g.y;
  g = L1[b + nNW]; a.x += g.w; a.y += g.z; a.z += g.y; a.w += g.x;
  g = L1[b + nNE]; a.x += g.y; a.y += g.w; a.z += g.x; a.w += g.z;
  return a;
}

__global__ __launch_bounds__(256)
void hdlut_kernel(const int* __restrict__ img, const float* __restrict__ w,
                  float* __restrict__ out) {
  __shared__ v4f lut[512];   // 8 KiB: [kt][idx] -> float4

  // Guaranteed-correct table fill (coalesced b128 LDS stores).
  {
    const v4f* wv = (const v4f*)w;
    lut[threadIdx.x]       = wv[threadIdx.x];
    lut[threadIdx.x + 256] = wv[threadIdx.x + 256];
  }

  // CDNA5 TDM: async tensor load of the same 8 KiB into the same LDS region
  // (idempotent with the fill above). One wave issues; EXEC ignored by TDM.
  if (threadIdx.x < 32) {
    unsigned long long ga = (unsigned long long)(uintptr_t)w;
    unsigned int lds_off =
        (unsigned int)(uintptr_t)(__attribute__((address_space(3))) void*)&lut[0];
    v4u g0;
    g0[0] = 1u;                                       // count=1, user mode
    g0[1] = lds_off;                                  // lds_addr (bytes)
    g0[2] = (unsigned int)(ga & 0xFFFFFFFFu);         // global_addr[31:0]
    g0[3] = (unsigned int)((ga >> 32) & 0x1FFFFFFu)   // global_addr[56:32]
          | (2u << 30);                               // type = 2 ("image")
    v8i g1;
    g1[0] = (int)(2u << 16);     // workgroup_mask=0, data_size=2 (4B)
    g1[1] = (int)(2048u << 16);  // tensor_dim0 = 2048 elements (bits 79:48 lo)
    g1[2] = (int)(1u << 16);     // tensor_dim0 hi=0, tensor_dim1 = 1
    g1[3] = (int)(2048u << 16);  // tensor_dim1 hi=0, tile_dim0 = 2048
    g1[4] = 1;                   // tile_dim1 = 1, tile_dim2 = 0
    g1[5] = 2048;                // tensor_dim0_stride[31:0]
    g1[6] = (int)(2048u << 16);  // stride0 hi=0, tensor_dim1_stride[15:0]
    g1[7] = 0;                   // tensor_dim1_stride[47:16]
    v4i z4; z4[0] = z4[1] = z4[2] = z4[3] = 0;
    v8i z8; z8[0]=z8[1]=z8[2]=z8[3]=z8[4]=z8[5]=z8[6]=z8[7]=0;
    __builtin_amdgcn_tensor_load_to_lds(g0, g1, z4, z4, z8, 0);
    __builtin_amdgcn_s_wait_tensorcnt(0);
  }
  __syncthreads();

  const int jt = blockIdx.x * blockDim.x + threadIdx.x; // 0..2047 (2 px each)
  const int j0 = jt * 2;
  const int i0 = blockIdx.y * 8;                        // 8 rows per thread

  // Column window j0-1 .. j0+2 with edge reflection.
  const int c0 = (j0 > 0) ? j0 - 1 : 1;
  const int c1 = j0;
  const int c2 = j0 + 1;
  const int c3 = (j0 + 2 < NN) ? j0 + 2 : NN - 2;

  int t0,t1,t2,t3, m0,m1,m2,m3, b0,b1,b2,b3;  // rolling 3-row window
  {
    const int rt = (i0 > 0) ? i0 - 1 : 1;
    const int* pt = img + (size_t)rt * NN;
    t0 = pt[c0]; t1 = pt[c1]; t2 = pt[c2]; t3 = pt[c3];
    const int* pm = img + (size_t)i0 * NN;
    m0 = pm[c0]; m1 = pm[c1]; m2 = pm[c2]; m3 = pm[c3];
  }

  for (int k = 0; k < 8; ++k) {
    const int i  = i0 + k;
    const int rb = (i + 1 < NN) ? i + 1 : NN - 2;
    const int* pb = img + (size_t)rb * NN;
    b0 = pb[c0]; b1 = pb[c1]; b2 = pb[c2]; b3 = pb[c3];

    // pixel A = (i, j0), pixel B = (i, j0+1)
    v4f A = gather8(lut, m1, m2, b1, m0, t1, b2, b0, t0, t2);
    v4f B = gather8(lut, m2, m3, b2, m1, t2, b3, b1, t1, t3);

    v4f r0, r1;
    r0.x = A.x * 0.5f; r0.y = A.y * 0.5f; r0.z = B.x * 0.5f; r0.w = B.y * 0.5f;
    r1.x = A.z * 0.5f; r1.y = A.w * 0.5f; r1.z = B.z * 0.5f; r1.w = B.w * 0.5f;

    const size_t o = (size_t)(2 * i) * ON + (size_t)(2 * j0);
    // Non-temporal: 256 MB write-only stream must not evict img from L2.
    __builtin_nontemporal_store(r0, (v4f*)(out + o));       // row 2i
    __builtin_nontemporal_store(r1, (v4f*)(out + o + ON));  // row 2i+1

    t0 = m0; t1 = m1; t2 = m2; t3 = m3;
    m0 = b0; m1 = b1; m2 = b2; m3 = b3;
  }
}

extern "C" void kernel_launch(void* const* d_in, const int* in_sizes, int n_in,
                              void* d_out, int out_size, void* d_ws, size_t ws_size,
                              hipStream_t stream) {
  (void)in_sizes; (void)n_in; (void)out_size; (void)d_ws; (void)ws_size;
  const int*   img = (const int*)d_in[0];     // (4096, 4096) int32
  const float* w   = (const float*)d_in[1];   // (2, 256, 4) float32
  float*       out = (float*)d_out;           // (8192, 8192) float32
  dim3 blk(256);
  dim3 grd(2048 / 256, 4096 / 8);             // 8 x 512 blocks
  hdlut_kernel<<<grd, blk, 0, stream>>>(img, w, out);
}